// MSEEigenRatioLoss_28484223107625
// MI455X (gfx1250) — compile-verified
//
#include <hip/hip_runtime.h>
#include <hip/hip_bf16.h>
#include <math.h>

// MI455X / gfx1250 implementation of the eigen-ratio MSE loss.
//
// Heavy phase: pairwise-distance top-16 per point, driven by
// V_WMMA_F32_16X16X4_F32 (exact f32 inner products, K padded 3->4).
// Inputs (~400KB) are L2/L1-resident vs 192MB L2 -> compute bound; budget
// goes to WMMA for the N^2 dot products and a compact register top-k.
// Selection uses a rolled candidate loop + branchless shift-insert so the
// hot loop stays instruction-cache resident (the round-1 fully-unrolled
// version was ~85KB of code).

typedef __attribute__((ext_vector_type(2))) float v2f;
typedef __attribute__((ext_vector_type(8))) float v8f;

#define NPTS 8192
#define KNN 16
#define CTILES (NPTS / 16)
#define ROW_STRIDE 17  // coprime with 64 LDS banks -> conflict-free row reads

__global__ __launch_bounds__(256) void eigen_ratio_kernel(
    const float* __restrict__ x, const float* __restrict__ y,
    float* __restrict__ ratio_out)
{
    // one 32x16 score slab per wave, padded rows
    __shared__ float smem[8][32 * ROW_STRIDE];

    const int wave = threadIdx.x >> 5;
    const int lane = threadIdx.x & 31;
    const int half = lane >> 4;   // 0: lanes 0-15, 1: lanes 16-31
    const int ln16 = lane & 15;

    // global wave id -> (tensor, batch, 32-row group)
    const int gwave = blockIdx.x * 8 + wave;   // 1024 waves total
    const int t   = gwave >> 9;                // 512 waves per tensor
    const int rem = gwave & 511;
    const int b   = rem >> 8;                  // 256 waves per batch
    const int rowbase = (rem & 255) * 32;

    const float* __restrict__ pts = (t == 0 ? x : y) + (size_t)b * NPTS * 3;
    float* __restrict__ rout = ratio_out + ((size_t)t * 2 + b) * NPTS;

    // ---- A tiles: two 16x4 f32 matrices (rows rowbase..+15, rowbase+16..+31)
    // ISA layout: lanes 0-15 hold K=0,1 across v0,v1; lanes 16-31 hold K=2,3.
    // K components are (px, py, pz, 0).
    v2f a0, a1;
    {
        const int r0 = rowbase + ln16;
        const int r1 = rowbase + 16 + ln16;
        if (half == 0) {
            a0.x = pts[r0 * 3 + 0]; a0.y = pts[r0 * 3 + 1];
            a1.x = pts[r1 * 3 + 0]; a1.y = pts[r1 * 3 + 1];
        } else {
            a0.x = pts[r0 * 3 + 2]; a0.y = 0.0f;
            a1.x = pts[r1 * 3 + 2]; a1.y = 0.0f;
        }
    }

    // ---- per-thread top-16, sorted ascending by s = |q|^2 - 2 p.q
    // (row-constant |p|^2 dropped: it cannot change per-row ordering)
    float bd[KNN];
    int   bi[KNN];
#pragma unroll
    for (int j = 0; j < KNN; ++j) { bd[j] = 3.4e38f; bi[j] = 0; }

    float* __restrict__ sm = &smem[wave][0];
    const int rowaddr = lane * ROW_STRIDE;

    for (int ct = 0; ct < CTILES; ++ct) {
        const int colbase = ct * 16;
        const int cpt = colbase + ln16;

        // prefetch next column tile (speculative, gfx1250 global_prefetch)
        if (ct + 1 < CTILES)
            __builtin_prefetch(&pts[(colbase + 16) * 3], 0, 1);

        // every lane loads full coords of its column point: gives both the
        // B-tile components and |q|^2 for its column of the score tile
        const float px = pts[cpt * 3 + 0];
        const float py = pts[cpt * 3 + 1];
        const float pz = pts[cpt * 3 + 2];
        const float sqc = px * px + py * py + pz * pz;

        // B tile 4x16: lanes 0-15 rows K=0,1 (x,y); lanes 16-31 rows K=2,3 (z,0)
        v2f bb;
        if (half == 0) { bb.x = px; bb.y = py; }
        else           { bb.x = pz; bb.y = 0.0f; }

        v8f acc0 = {};
        v8f acc1 = {};
        acc0 = __builtin_amdgcn_wmma_f32_16x16x4_f32(
            false, a0, false, bb, (short)0, acc0, false, false);
        acc1 = __builtin_amdgcn_wmma_f32_16x16x4_f32(
            false, a1, false, bb, (short)0, acc1, false, false);

        // C/D layout: VGPR r, lane l -> row = r + 8*(l>=16), col = l&15
#pragma unroll
        for (int r = 0; r < 8; ++r) {
            sm[(half * 8 + r) * ROW_STRIDE + ln16]      = sqc - 2.0f * acc0[r];
            sm[(16 + half * 8 + r) * ROW_STRIDE + ln16] = sqc - 2.0f * acc1[r];
        }
        // cross-lane LDS write -> read within the wave; LDS is in-order per
        // wave but make the dependence explicit with the CDNA5 split counter
        asm volatile("s_wait_dscnt 0" ::: "memory");

        // lane l owns row (rowbase + l). Rolled loop: the insert chain exists
        // once in the binary, keeping the hot loop I$-resident.
#pragma unroll 1
        for (int c = 0; c < 16; ++c) {
            const float s = sm[rowaddr + c];
            if (s < bd[KNN - 1]) {            // 1-compare fast reject
                const int ci = colbase + c;
                // branchless shift-insert, slots processed descending so
                // each step reads only not-yet-updated (old) entries
#pragma unroll
                for (int j = KNN - 1; j >= 1; --j) {
                    const bool cj   = s < bd[j];
                    const bool cjm1 = s < bd[j - 1];
                    const float nd  = cjm1 ? bd[j - 1] : s;
                    const int   ni  = cjm1 ? bi[j - 1] : ci;
                    bd[j] = cj ? nd : bd[j];
                    bi[j] = cj ? ni : bi[j];
                }
                const bool c0 = s < bd[0];
                bd[0] = c0 ? s  : bd[0];
                bi[0] = c0 ? ci : bi[0];
            }
        }
    }

    // ---- neighborhood covariance: cov = E[q q^T] - m m^T, k = 16
    float sx = 0, sy = 0, sz = 0;
    float sxx = 0, syy = 0, szz = 0, sxy = 0, sxz = 0, syz = 0;
#pragma unroll
    for (int j = 0; j < KNN; ++j) {
        const int n = bi[j];
        const float qx = pts[n * 3 + 0];
        const float qy = pts[n * 3 + 1];
        const float qz = pts[n * 3 + 2];
        sx += qx; sy += qy; sz += qz;
        sxx += qx * qx; syy += qy * qy; szz += qz * qz;
        sxy += qx * qy; sxz += qx * qz; syz += qy * qz;
    }
    const float ik = 1.0f / (float)KNN;
    const float mx = sx * ik, my = sy * ik, mz = sz * ik;
    const float cxx = sxx * ik - mx * mx;
    const float cyy = syy * ik - my * my;
    const float czz = szz * ik - mz * mz;
    const float cxy = sxy * ik - mx * my;
    const float cxz = sxz * ik - mx * mz;
    const float cyz = syz * ik - my * mz;

    // ---- closed-form eigenvalues of a symmetric 3x3 (trigonometric method)
    const float q3 = (cxx + cyy + czz) * (1.0f / 3.0f);
    const float p1 = cxy * cxy + cxz * cxz + cyz * cyz;
    const float aq = cxx - q3, bq = cyy - q3, cq = czz - q3;
    const float p2 = aq * aq + bq * bq + cq * cq + 2.0f * p1;
    const float p  = sqrtf(p2 * (1.0f / 6.0f));

    float ratio;
    if (p < 1e-20f) {
        ratio = 1.0f;  // isotropic / degenerate neighborhood
    } else {
        const float invp = 1.0f / p;
        const float b00 = aq * invp, b11 = bq * invp, b22 = cq * invp;
        const float b01 = cxy * invp, b02 = cxz * invp, b12 = cyz * invp;
        float detB = b00 * (b11 * b22 - b12 * b12)
                   - b01 * (b01 * b22 - b12 * b02)
                   + b02 * (b01 * b12 - b11 * b02);
        float r = 0.5f * detB;
        r = fminf(1.0f, fmaxf(-1.0f, r));
        const float phi = acosf(r) * (1.0f / 3.0f);
        const float e1 = q3 + 2.0f * p * cosf(phi);                       // lambda_max
        const float e3 = q3 + 2.0f * p * cosf(phi + 2.0943951023931953f); // lambda_min
        const float e2 = 3.0f * q3 - e1 - e3;                             // lambda_mid
        ratio = e1 / e2;
    }

    rout[rowbase + lane] = ratio;
}

__global__ __launch_bounds__(256) void mse_reduce_kernel(
    const float* __restrict__ ratios, float* __restrict__ out)
{
    __shared__ float red[256];
    float acc = 0.0f;
    for (int i = threadIdx.x; i < 2 * NPTS; i += 256) {
        const float d = ratios[i] - ratios[2 * NPTS + i];
        acc += d * d;
    }
    red[threadIdx.x] = acc;
    __syncthreads();
#pragma unroll
    for (int s = 128; s > 0; s >>= 1) {
        if (threadIdx.x < (unsigned)s) red[threadIdx.x] += red[threadIdx.x + s];
        __syncthreads();
    }
    if (threadIdx.x == 0) out[0] = red[0] * (1.0f / (float)(2 * NPTS));
}

extern "C" void kernel_launch(void* const* d_in, const int* in_sizes, int n_in,
                              void* d_out, int out_size, void* d_ws, size_t ws_size,
                              hipStream_t stream) {
    const float* x = (const float*)d_in[0];   // [2, 8192, 3] f32
    const float* y = (const float*)d_in[1];   // [2, 8192, 3] f32
    float* ratios = (float*)d_ws;             // [2 tensors][2 batches][8192] f32
    float* out = (float*)d_out;               // scalar f32

    // 1024 waves = 2 tensors * 2 batches * 256 row-groups of 32 points
    eigen_ratio_kernel<<<128, 256, 0, stream>>>(x, y, ratios);
    mse_reduce_kernel<<<1, 256, 0, stream>>>(ratios, out);
}